// PillarFeatureGenerator_10093173145812
// MI455X (gfx1250) — compile-verified
//
#include <hip/hip_runtime.h>
#include <hip/hip_bf16.h>

#define PW   0.16f
#define PH   0.16f
#define XMN  0.0f
#define YMN  (-40.0f)
#define NXC  440
#define NYC  500
#define GC   220000
#define MAXP 32
#define NBLK 860   /* ceil(GC/256) */

// native clang vector type (accepted by __builtin_nontemporal_store)
typedef __attribute__((ext_vector_type(4))) float v4f;

// per-sample scratch slab layout (bytes)
#define OFF_COUNTS  ((size_t)0)
#define OFF_SUMS    ((size_t)4  * GC)           // 3 floats / cell
#define OFF_SLOT    ((size_t)16 * GC)
#define OFF_IDX     ((size_t)20 * GC)           // 32 ints / cell
#define OFF_PART    ((size_t)148 * GC)
#define SLAB_BYTES  ((((size_t)148 * GC + 4 * NBLK) + 255) & ~(size_t)255)

// ---------------------------------------------------------------- fill kernel
// Streaming constant fill with non-temporal (TH=NT) b128 stores: the 508 MB
// output region is write-once and must not evict the ~70 MB of hot scratch
// (counts/sums/idxbuf) or the 3.8 MB point cloud from the 192 MB L2.
__global__ void __launch_bounds__(256) k_fill(v4f* __restrict__ dst,
                                              long long n4, float value) {
    long long i = (long long)blockIdx.x * blockDim.x + threadIdx.x;
    const long long stride = (long long)gridDim.x * blockDim.x;
    const v4f v = {value, value, value, value};
    for (; i < n4; i += stride)
        __builtin_nontemporal_store(v, dst + i);
}

__global__ void __launch_bounds__(256) k_init_ws(char* ws, int n4) {
    v4f* p = (v4f*)(ws + (size_t)blockIdx.y * SLAB_BYTES);
    int i = blockIdx.x * 256 + threadIdx.x;
    if (i < n4) {
        const v4f z = {0.f, 0.f, 0.f, 0.f};
        p[i] = z;
    }
}

// ---------------------------------------------------------------- point pass
// CDNA5 async copy: stage this block's 256 points (4 KB) into LDS with the
// async-to-LDS DMA path (ASYNCcnt), then consume from LDS.
__global__ void __launch_bounds__(256) k_points(const float4* __restrict__ pc,
                                                char* __restrict__ ws, int B, int N) {
    __shared__ float4 tile[256];
    const int total = B * N;
    const int p = blockIdx.x * 256 + threadIdx.x;
    const int safe = (p < total) ? p : 0;

    unsigned ldsAddr = (unsigned)(size_t)(&tile[threadIdx.x]);
    unsigned long long gaddr = (unsigned long long)(size_t)(pc + safe);
    // speculative prefetch of the next tile (dropped if OOB)
    __builtin_prefetch(pc + ((safe + 256 < total) ? safe + 256 : safe), 0, 0);
    asm volatile("global_load_async_to_lds_b128 %0, %1, off"
                 :: "v"(ldsAddr), "v"(gaddr) : "memory");
    asm volatile("s_wait_asynccnt 0" ::: "memory");
    const float4 pt = tile[threadIdx.x];

    if (p >= total) return;
    const int b = p / N;
    const int n = p - b * N;
    char* slab   = ws + (size_t)b * SLAB_BYTES;
    int*   counts = (int*)  (slab + OFF_COUNTS);
    float* sums   = (float*)(slab + OFF_SUMS);
    int*   idxb   = (int*)  (slab + OFF_IDX);

    // match reference exactly: floor((x-min)/w), clamp to NX-1 (IEEE f32 div)
    float fx = fminf(floorf((pt.x - XMN) / PW), (float)(NXC - 1));
    float fy = fminf(floorf((pt.y - YMN) / PH), (float)(NYC - 1));
    int cell = (int)fy * NXC + (int)fx;

    int s = atomicAdd(&counts[cell], 1);          // count + provisional slot
    if (s < MAXP) idxb[(size_t)cell * MAXP + s] = n;
    unsafeAtomicAdd(&sums[cell * 3 + 0], pt.x);   // global_atomic_add_f32
    unsafeAtomicAdd(&sums[cell * 3 + 1], pt.y);
    unsafeAtomicAdd(&sums[cell * 3 + 2], pt.z);
}

// ---------------------------------------------------------------- scan stage 1
__global__ void __launch_bounds__(256) k_block_count(char* ws) {
    const int b = blockIdx.y;
    char* slab = ws + (size_t)b * SLAB_BYTES;
    const int* counts = (const int*)(slab + OFF_COUNTS);
    int* partials = (int*)(slab + OFF_PART);
    const int g = blockIdx.x * 256 + threadIdx.x;
    __shared__ int red[256];
    red[threadIdx.x] = (g < GC && counts[g] > 0) ? 1 : 0;
    __syncthreads();
    for (int s = 128; s > 0; s >>= 1) {
        if (threadIdx.x < s) red[threadIdx.x] += red[threadIdx.x + s];
        __syncthreads();
    }
    if (threadIdx.x == 0) partials[blockIdx.x] = red[0];
}

// ---------------------------------------------------------------- scan stage 2
__global__ void __launch_bounds__(1024) k_scan_partials(char* ws) {
    char* slab = ws + (size_t)blockIdx.x * SLAB_BYTES;
    int* partials = (int*)(slab + OFF_PART);
    __shared__ int sh[1024];
    const int t = threadIdx.x;
    const int v = (t < NBLK) ? partials[t] : 0;
    sh[t] = v;
    __syncthreads();
    for (int off = 1; off < 1024; off <<= 1) {
        int add = (t >= off) ? sh[t - off] : 0;
        __syncthreads();
        sh[t] += add;
        __syncthreads();
    }
    if (t < NBLK) partials[t] = sh[t] - v;   // exclusive base per block
}

// ---------------------------------------------------------------- scan stage 3
__global__ void __launch_bounds__(256) k_slot(char* ws) {
    const int b = blockIdx.y;
    char* slab = ws + (size_t)b * SLAB_BYTES;
    const int* counts   = (const int*)(slab + OFF_COUNTS);
    const int* partials = (const int*)(slab + OFF_PART);
    int* slot = (int*)(slab + OFF_SLOT);
    const int g = blockIdx.x * 256 + threadIdx.x;
    const int t = threadIdx.x;
    __shared__ int sh[256];
    sh[t] = (g < GC && counts[g] > 0) ? 1 : 0;
    __syncthreads();
    for (int off = 1; off < 256; off <<= 1) {
        int add = (t >= off) ? sh[t - off] : 0;
        __syncthreads();
        sh[t] += add;
        __syncthreads();
    }
    if (g < GC) slot[g] = partials[blockIdx.x] + sh[t] - 1;
}

// ---------------------------------------------------------------- emit
__global__ void __launch_bounds__(256) k_emit(const float4* __restrict__ pc,
                                              char* __restrict__ ws,
                                              float* __restrict__ out,
                                              float* __restrict__ pids,
                                              int B, int N) {
    const int t = blockIdx.x * 256 + threadIdx.x;
    if (t >= B * GC) return;
    const int b = t / GC;
    const int g = t - b * GC;
    char* slab = ws + (size_t)b * SLAB_BYTES;
    const int*   counts = (const int*)  (slab + OFF_COUNTS);
    const float* sums   = (const float*)(slab + OFF_SUMS);
    const int*   slot   = (const int*)  (slab + OFF_SLOT);
    const int*   idxb   = (const int*)  (slab + OFF_IDX);

    const int c = counts[g];
    if (c == 0) return;
    const int k = (c < MAXP) ? c : MAXP;

    int idxs[MAXP];
    const int* buf = idxb + (size_t)g * MAXP;
    for (int r = 0; r < k; ++r) idxs[r] = buf[r];
    for (int i = 1; i < k; ++i) {              // stable rank = ascending index
        int v = idxs[i], j = i - 1;
        while (j >= 0 && idxs[j] > v) { idxs[j + 1] = idxs[j]; --j; }
        idxs[j + 1] = v;
    }

    const float inv = 1.0f / (float)c;
    const float mx = sums[g * 3 + 0] * inv;
    const float my = sums[g * 3 + 1] * inv;
    const float mz = sums[g * 3 + 2] * inv;

    const int ixg = g % NXC, iyg = g / NXC;
    const float cx = XMN + ((float)ixg) * PW + PW * 0.5f;
    const float cy = YMN + ((float)iyg) * PH + PH * 0.5f;

    const int sl = slot[g];
    pids[(size_t)b * GC + sl] = (float)g;

    float* orow = out + ((size_t)b * GC + (size_t)sl) * (size_t)(MAXP * 9);
    for (int r = 0; r < k; ++r) {
        const float4 pt = pc[(size_t)b * N + idxs[r]];
        float* o = orow + r * 9;
        o[0] = pt.x; o[1] = pt.y; o[2] = pt.z; o[3] = pt.w;
        o[4] = fabsf(pt.x - mx);
        o[5] = fabsf(pt.y - my);
        o[6] = fabsf(pt.z - mz);
        o[7] = cx - pt.x;
        o[8] = cy - pt.y;
    }
}

// ---------------------------------------------------------------- launch
extern "C" void kernel_launch(void* const* d_in, const int* in_sizes, int n_in,
                              void* d_out, int out_size, void* d_ws, size_t ws_size,
                              hipStream_t stream) {
    const float4* pc = (const float4*)d_in[0];
    float* out = (float*)d_out;
    char* ws = (char*)d_ws;

    const int B = out_size / (GC * 289);            // G*32*9 + G per sample
    const int N = in_sizes[0] / (4 * B);
    float* pids = out + (size_t)B * GC * (MAXP * 9);

    const long long feat4 = (long long)B * GC * (MAXP * 9) / 4;   // zeros region
    const long long pid4  = (long long)out_size / 4 - feat4;      // -1 region

    // split fills: constant value stays in registers (no scratch cndmask),
    // non-temporal b128 stores keep the 508 MB stream out of L2
    k_fill<<<32768, 256, 0, stream>>>((v4f*)out, feat4, 0.0f);
    k_fill<<<2048, 256, 0, stream>>>((v4f*)out + feat4, pid4, -1.0f);
    k_init_ws<<<dim3((GC + 255) / 256, B), 256, 0, stream>>>(ws, GC); // 16*GC bytes
    k_points<<<(B * N + 255) / 256, 256, 0, stream>>>(pc, ws, B, N);
    k_block_count<<<dim3(NBLK, B), 256, 0, stream>>>(ws);
    k_scan_partials<<<B, 1024, 0, stream>>>(ws);
    k_slot<<<dim3(NBLK, B), 256, 0, stream>>>(ws);
    k_emit<<<(B * GC + 255) / 256, 256, 0, stream>>>(pc, ws, out, pids, B, N);
}